// ECELoss_10531259810371
// MI455X (gfx1250) — compile-verified
//
#include <hip/hip_runtime.h>
#include <stdint.h>

#define ECE_NBINS 15
#define ECE_C 64
#define ROWS_PER_STAGE 16
#define WAVES_PER_WG 8
#define WG_THREADS 256
#define NUM_WGS 2048
#define STAGE_BYTES (ROWS_PER_STAGE * ECE_C * 4) /* 4096 B per buffer */
#define STAGE_F4 (ROWS_PER_STAGE * ECE_C / 4)    /* 256 float4 per buffer */

// ---- CDNA5 async DMA: copy one 4KB stage (16 rows x 256B) global -> LDS.
// 8 x global_load_async_to_lds_b128: each instruction moves 32 lanes x 16B = 512B,
// fully contiguous in both global memory and LDS. Tracked by ASYNCcnt.
// th:TH_LOAD_NT — pure streaming data (1 GB read once), don't rinse L2 with it.
__device__ __forceinline__ void async_stage_load(unsigned ldsAddr,
                                                 unsigned long long base,
                                                 unsigned gvoff) {
#pragma unroll
  for (int k = 0; k < 8; ++k) {
    asm volatile("global_load_async_to_lds_b128 %0, %1, %2 th:TH_LOAD_NT"
                 :: "v"(ldsAddr + (unsigned)(k * 512)),
                    "v"(gvoff + (unsigned)(k * 512)),
                    "s"(base)
                 : "memory");
  }
}

__global__ void __launch_bounds__(WG_THREADS)
ece_main(const float* __restrict__ softmaxes, const int* __restrict__ labels,
         float* __restrict__ wsf /* conf_sum[15], acc_sum[15] */,
         unsigned* __restrict__ wsc /* cnt[15] */, int nstages) {
  __shared__ float4 sbuf[WAVES_PER_WG][2][STAGE_F4];
  __shared__ float s_conf[ECE_NBINS];
  __shared__ float s_acc[ECE_NBINS];
  __shared__ unsigned s_cnt[ECE_NBINS];

  const int tid = threadIdx.x;
  if (tid < ECE_NBINS) { s_conf[tid] = 0.f; s_acc[tid] = 0.f; s_cnt[tid] = 0u; }
  __syncthreads();

  const int wave = tid >> 5;
  const int lane = tid & 31;
  const int half = lane >> 4;  // 0: even row of pair, 1: odd row
  const int lh   = lane & 15;  // position within half-wave

  // LDS byte addresses (low 32 bits of the generic pointer = workgroup-relative
  // LDS offset, which is what the async instruction's VDST expects).
  unsigned ldsB[2];
  ldsB[0] = (unsigned)(uintptr_t)(&sbuf[wave][0][0]) + (unsigned)lane * 16u;
  ldsB[1] = (unsigned)(uintptr_t)(&sbuf[wave][1][0]) + (unsigned)lane * 16u;
  const unsigned long long base = (unsigned long long)(uintptr_t)softmaxes;
  const unsigned laneByte = (unsigned)lane * 16u;

  int s = blockIdx.x * WAVES_PER_WG + wave;     // stage index (uniform per wave)
  const int stride = gridDim.x * WAVES_PER_WG;

  int cur = 0;
  if (s < nstages)
    async_stage_load(ldsB[0], base, (unsigned)s * (unsigned)STAGE_BYTES + laneByte);

  while (s < nstages) {
    const int snext = s + stride;

    // Labels for this stage's 16 rows: independent of the LDS tile, so load
    // BEFORE the async wait to overlap its latency with the DMA completion.
    const int r0 = s * ROWS_PER_STAGE;
    const int mylab = labels[r0 + lh];  // lanes 0..15 hold labels[r0+lh] (mirrored 16..31)

    if (snext < nstages) {
      async_stage_load(ldsB[cur ^ 1], base,
                       (unsigned)snext * (unsigned)STAGE_BYTES + laneByte);
      asm volatile("s_wait_asynccnt 0x8" ::: "memory"); // current stage complete
    } else {
      asm volatile("s_wait_asynccnt 0x0" ::: "memory");
    }

    const float4* buf = &sbuf[wave][cur][0];

#pragma unroll
    for (int j = 0; j < 8; ++j) {
      const int row = 2 * j + half;
      float4 v = buf[row * 16 + lh];      // ds_load_b128, conflict-free (2x64 banks)
      float m = v.x; int mi = 4 * lh;
      if (v.y > m) { m = v.y; mi = 4 * lh + 1; }
      if (v.z > m) { m = v.z; mi = 4 * lh + 2; }
      if (v.w > m) { m = v.w; mi = 4 * lh + 3; }
      // Butterfly over 16 lanes of each half: lexicographic max on (value, -index)
      // => global max with first-occurrence index, matching jnp.argmax.
#pragma unroll
      for (int d = 1; d < 16; d <<= 1) {
        float om = __shfl_xor(m, d, 32);
        int   oi = __shfl_xor(mi, d, 32);
        if (om > m || (om == m && oi < mi)) { m = om; mi = oi; }
      }
      const int lab = __shfl(mylab, 2 * j + half, 32); // convergent shuffle
      if (lh == 0) {
        const float conf = m;
        const float acc  = (mi == lab) ? 1.0f : 0.0f;
        // searchsorted(linspace(0,1,16), conf, 'left') - 1 ;  conf > 0 always
        int bid = 0;
#pragma unroll
        for (int b = 1; b <= 14; ++b)
          bid += (conf > (float)b * (1.0f / 15.0f)) ? 1 : 0;
        atomicAdd(&s_cnt[bid], 1u);
        atomicAdd(&s_conf[bid], conf);
        atomicAdd(&s_acc[bid], acc);
      }
    }
    s = snext;
    cur ^= 1;
  }

  __syncthreads();
  if (tid < ECE_NBINS) {
    atomicAdd(&wsc[tid], s_cnt[tid]);
    atomicAdd(&wsf[tid], s_conf[tid]);
    atomicAdd(&wsf[ECE_NBINS + tid], s_acc[tid]);
  }
}

__global__ void ece_init(float* wsf, unsigned* wsc) {
  const int t = threadIdx.x;
  if (t < 2 * ECE_NBINS) wsf[t] = 0.f;
  if (t < ECE_NBINS) wsc[t] = 0u;
}

__global__ void ece_final(const float* __restrict__ wsf,
                          const unsigned* __restrict__ wsc,
                          float* __restrict__ out, int n) {
  if (threadIdx.x == 0) {
    float total = 0.f;
    const float invN = 1.0f / (float)n;
    for (int b = 0; b < ECE_NBINS; ++b) {
      const float cnt = (float)wsc[b];
      const float denom = fmaxf(cnt, 1.0f);
      const float avg_conf = wsf[b] / denom;
      const float avg_acc  = wsf[ECE_NBINS + b] / denom;
      total += (cnt > 0.f) ? fabsf(avg_conf - avg_acc) * cnt * invN : 0.f;
    }
    out[0] = total;
  }
}

extern "C" void kernel_launch(void* const* d_in, const int* in_sizes, int n_in,
                              void* d_out, int out_size, void* d_ws, size_t ws_size,
                              hipStream_t stream) {
  (void)n_in; (void)out_size; (void)ws_size;
  const float* softmaxes = (const float*)d_in[0];
  const int*   labels    = (const int*)d_in[1];
  const int n = in_sizes[1];                 // number of rows (N)
  const int nstages = n / ROWS_PER_STAGE;    // N divisible by 16 for this problem

  float*    wsf = (float*)d_ws;                                      // 30 floats
  unsigned* wsc = (unsigned*)((char*)d_ws + 2 * ECE_NBINS * sizeof(float)); // 15 u32

  ece_init<<<1, 64, 0, stream>>>(wsf, wsc);
  ece_main<<<NUM_WGS, WG_THREADS, 0, stream>>>(softmaxes, labels, wsf, wsc, nstages);
  ece_final<<<1, 32, 0, stream>>>(wsf, wsc, (float*)d_out, n);
}